// SingleColumnModel_33036888441441
// MI455X (gfx1250) — compile-verified
//
#include <hip/hip_runtime.h>
#include <cstdint>
#include <cstddef>

#ifndef __has_builtin
#define __has_builtin(x) 0
#endif

// ---- problem constants (match reference) ----
namespace {
constexpr int   kB     = 2048;
constexpr int   kNZ    = 128;
constexpr int   kNT    = 400;
constexpr int   kNOut  = 40;
constexpr int   kNSnap = kNT / kNOut;   // 10
constexpr float kDT    = 50.0f;
constexpr float kFcor  = 1.0e-4f;
constexpr float kTflxS = -5.0e-5f;
constexpr float kSflxS =  1.0e-6f;
constexpr float kTflxB =  0.0f;
constexpr float kRflxS =  3.7e-5f;
constexpr float kUstrS =  1.0e-4f;
constexpr float kVstrS =  0.0f;
constexpr float kCp    = 3985.0f;

constexpr int COLS   = 32;          // one wave32 per block, one column per lane
constexpr int STRIDE = kNZ + 1;     // 129 -> lane l hits bank (l+k)%64: conflict-free
constexpr int NARR   = 11;          // t,s,u,v, a_t,inv_t,cp_t, a_m,inv_m,cp_m, frc_t
}

// ---- CDNA5 async global->LDS staging (ASYNCcnt path), guarded fallbacks ----
__device__ __forceinline__ void async_copy_b32(const float* g, float* l) {
#if __has_builtin(__builtin_amdgcn_global_load_async_to_lds_b32)
  __builtin_amdgcn_global_load_async_to_lds_b32(
      (__attribute__((address_space(1))) int*)(g),
      (__attribute__((address_space(3))) int*)(l),
      /*offset=*/0, /*cpol=*/0);
#else
  *l = *g;
#endif
}

__device__ __forceinline__ void wait_async0() {
#if __has_builtin(__builtin_amdgcn_s_wait_asynccnt)
  __builtin_amdgcn_s_wait_asynccnt(0);
#else
  asm volatile("s_wait_asynccnt 0x0" ::: "memory");
#endif
}

__device__ __forceinline__ float swfrac_flux(float zw) {
  // RFLX_SFC * (0.58*exp(z/0.35) + 0.42*exp(z/23))
  return kRflxS * (0.58f * expf(zw * (1.0f / 0.35f)) +
                   0.42f * expf(zw * (1.0f / 23.0f)));
}

// One-time LU factorization of (I - dt*D) for diffusivity K at w-points.
__device__ __forceinline__ void factor_column(const float* __restrict__ K,
                                              size_t kbase,
                                              const float* __restrict__ hzp,
                                              float* Pa, float* Pi, float* Pc) {
  float cp_prev = 0.0f;
  for (int k = 0; k < kNZ; ++k) {
    const float hzk = hzp[k];
    float a = 0.0f, c = 0.0f;
    if (k > 0)
      a = -kDT * K[kbase + k]     / (hzk * 0.5f * (hzp[k - 1] + hzk));
    if (k < kNZ - 1)
      c = -kDT * K[kbase + k + 1] / (hzk * 0.5f * (hzk + hzp[k + 1]));
    const float b    = 1.0f - a - c;
    const float den  = b - a * cp_prev;
    const float invd = 1.0f / den;
    const float cp   = c * invd;
    Pa[k] = a;  Pi[k] = invd;  Pc[k] = cp;
    cp_prev = cp;
  }
}

__global__ __launch_bounds__(COLS) void scm_timestep_kernel(
    const float* __restrict__ u0,  const float* __restrict__ v0,
    const float* __restrict__ t0,  const float* __restrict__ s0,
    const float* __restrict__ akv, const float* __restrict__ akt,
    const float* __restrict__ eps, const float* __restrict__ hz,
    float* __restrict__ out)
{
  __shared__ float smem[NARR * COLS * STRIDE];  // ~177 KB of the 320 KB WGP LDS
  __shared__ float shz[kNZ];

  const int tid = threadIdx.x;
  const int col = blockIdx.x * COLS + tid;
  if (col >= kB) return;

  float* Pt  = &smem[(0  * COLS + tid) * STRIDE];
  float* Ps  = &smem[(1  * COLS + tid) * STRIDE];
  float* Pu  = &smem[(2  * COLS + tid) * STRIDE];
  float* Pv  = &smem[(3  * COLS + tid) * STRIDE];
  float* Pat = &smem[(4  * COLS + tid) * STRIDE];
  float* Pit = &smem[(5  * COLS + tid) * STRIDE];
  float* Pct = &smem[(6  * COLS + tid) * STRIDE];
  float* Pam = &smem[(7  * COLS + tid) * STRIDE];
  float* Pim = &smem[(8  * COLS + tid) * STRIDE];
  float* Pcm = &smem[(9  * COLS + tid) * STRIDE];
  float* Pfr = &smem[(10 * COLS + tid) * STRIDE];

  // 1) Kick off async staging of the initial state (overlaps with precompute).
  {
    const size_t base = (size_t)col * kNZ;
    for (int k = 0; k < kNZ; ++k) {
      async_copy_b32(&u0[base + k], &Pu[k]);
      async_copy_b32(&v0[base + k], &Pv[k]);
      async_copy_b32(&t0[base + k], &Pt[k]);
      async_copy_b32(&s0[base + k], &Ps[k]);
    }
  }

  // 2) Cooperative hz load (uniform across columns -> broadcast reads later).
  for (int k = tid; k < kNZ; k += COLS) shz[k] = hz[k];
  __syncthreads();

  // 3) One-time precompute: forcing + LU factors (hidden under the async DMA).
  {
    float hsum = 0.0f;
    for (int k = 0; k < kNZ; ++k) hsum += shz[k];
    const size_t ebase = (size_t)col * (kNZ + 1);
    float zw  = -hsum;
    float fco = swfrac_flux(zw);
    for (int k = 0; k < kNZ; ++k) {
      zw += shz[k];
      const float fcn = swfrac_flux(zw);
      float dv = (fcn - fco) / shz[k];
      if (k == kNZ - 1) dv += kTflxS / shz[k];
      if (k == 0)       dv -= kTflxB / shz[k];
      const float ec = 0.5f * (eps[ebase + k] + eps[ebase + k + 1]);
      Pfr[k] = kDT * (dv + ec / kCp);
      fco = fcn;
    }
    factor_column(akt, ebase, shz, Pat, Pit, Pct);  // tracer matrix
    factor_column(akv, ebase, shz, Pam, Pim, Pcm);  // momentum matrix
  }

  // 4) State must be resident before the time loop.
  wait_async0();
  __syncthreads();

  // 5) Time stepping: 400 steps, all 4 solves fused (4 independent chains).
  const float gc    = kDT * kFcor;
  const float rot   = 1.0f / (1.0f + gc * gc);
  const float hzs   = shz[kNZ - 1];
  const float frcS  = kDT * kSflxS / hzs;   // salinity surface flux
  const float frcU  = kDT * kUstrS / hzs;   // wind stress u
  const float frcV  = kDT * kVstrS / hzs;   // wind stress v (0)

  for (int it = 0; it < kNT; ++it) {
    if ((it % kNOut) == 0) {                 // snapshot BEFORE the block
      const int snap = it / kNOut;
      float* ou = out + (((size_t)(0 * kNSnap + snap) * kB + col) * kNZ);
      float* ov = out + (((size_t)(1 * kNSnap + snap) * kB + col) * kNZ);
      float* ot = out + (((size_t)(2 * kNSnap + snap) * kB + col) * kNZ);
      float* os = out + (((size_t)(3 * kNSnap + snap) * kB + col) * kNZ);
      for (int k = 0; k < kNZ; ++k) {
        ou[k] = Pu[k]; ov[k] = Pv[k]; ot[k] = Pt[k]; os[k] = Ps[k];
      }
    }

    // Forward sweep: Coriolis + forcing + dp; dp overwrites state in place.
    float dpt = 0.0f, dps = 0.0f, dpu = 0.0f, dpv = 0.0f;
#pragma unroll 4
    for (int k = 0; k < kNZ; ++k) {
      const float a_t = Pat[k], i_t = Pit[k];
      const float a_m = Pam[k], i_m = Pim[k];
      float dT = Pt[k] + Pfr[k];
      float dS = Ps[k];
      const float uo = Pu[k], vo = Pv[k];
      float u1 = (uo + gc * vo) * rot;
      float v1 = (vo - gc * uo) * rot;
      if (k == kNZ - 1) { dS += frcS; u1 += frcU; v1 += frcV; }
      dpt = (dT - a_t * dpt) * i_t;
      dps = (dS - a_t * dps) * i_t;
      dpu = (u1 - a_m * dpu) * i_m;
      dpv = (v1 - a_m * dpv) * i_m;
      Pt[k] = dpt; Ps[k] = dps; Pu[k] = dpu; Pv[k] = dpv;
    }

    // Backward substitution: x_k = dp_k - cp_k * x_{k+1}; overwrites dp -> state.
    float xt = 0.0f, xs = 0.0f, xu = 0.0f, xv = 0.0f;
#pragma unroll 4
    for (int k = kNZ - 1; k >= 0; --k) {
      const float ct = Pct[k], cm = Pcm[k];
      xt = Pt[k] - ct * xt;
      xs = Ps[k] - ct * xs;
      xu = Pu[k] - cm * xu;
      xv = Pv[k] - cm * xv;
      Pt[k] = xt; Ps[k] = xs; Pu[k] = xu; Pv[k] = xv;
    }
  }
}

extern "C" void kernel_launch(void* const* d_in, const int* in_sizes, int n_in,
                              void* d_out, int out_size, void* d_ws, size_t ws_size,
                              hipStream_t stream) {
  const float* u   = (const float*)d_in[0];
  const float* v   = (const float*)d_in[1];
  const float* t   = (const float*)d_in[2];
  const float* s   = (const float*)d_in[3];
  const float* akv = (const float*)d_in[4];
  const float* akt = (const float*)d_in[5];
  const float* eps = (const float*)d_in[6];
  const float* hz  = (const float*)d_in[7];
  float* out = (float*)d_out;

  dim3 grid(kB / COLS), block(COLS);
  scm_timestep_kernel<<<grid, block, 0, stream>>>(u, v, t, s, akv, akt, eps, hz, out);

  (void)in_sizes; (void)n_in; (void)out_size; (void)d_ws; (void)ws_size;
}